// cmdnet_bin_36936718745672
// MI455X (gfx1250) — compile-verified
//
#include <hip/hip_runtime.h>

typedef __attribute__((ext_vector_type(2))) float v2f;
typedef __attribute__((ext_vector_type(8))) float v8f;

#define NRr 128
#define NTt 64
#define NITER 64

// LDS row strides chosen for 64-bank conflict avoidance:
//  HT_STRIDE 72: rows k and k+2 differ by 16 banks -> WMMA operand half-waves disjoint
//  HH_STRIDE 68: rows v and v+8 differ by 32 banks -> WMMA C-store half-waves disjoint
#define HT_STRIDE 72
#define HH_STRIDE 68

__device__ __forceinline__ float fast_tanh(float x) {
  x = fminf(fmaxf(x, -15.0f), 15.0f);
  // tanh(x) = (e^{2x} - 1) / (e^{2x} + 1);  e^{2x} = exp2(x * 2*log2(e))
  float e = __builtin_amdgcn_exp2f(x * 2.8853900817779268f);
  return (e - 1.0f) * __builtin_amdgcn_rcpf(e + 1.0f);
}

__global__ __launch_bounds__(256) void cmdnet_kernel(
    const float* __restrict__ yt, const float* __restrict__ Ht,
    const float* __restrict__ sigmat0, const float* __restrict__ taui,
    const float* __restrict__ delta, float* __restrict__ out, int B) {
  __shared__ __align__(16) float sHt[NRr * HT_STRIDE];  // 36 KB
  __shared__ float sHH[NTt * HH_STRIDE];                // 17 KB
  __shared__ float sYt[NRr];
  __shared__ float sXt[NTt];

  const int b = blockIdx.x;
  const int tid = (int)threadIdx.x;
  const int lane = tid & 31;
  const int wave = tid >> 5;

  // ---- stage Ht[b] (128x64 f32, row-major) into padded LDS ----
  const float4* gHt = (const float4*)(Ht + (size_t)b * (NRr * NTt));
  for (int u = tid; u < (NRr * NTt) / 4; u += 256) {
    float4 v = gHt[u];
    int idx = u * 4;
    int r = idx >> 6;       // row (receive antenna)
    int c = idx & 63;       // col (transmit antenna)
    *(float4*)(&sHt[r * HT_STRIDE + c]) = v;
  }
  if (tid < NRr) sYt[tid] = yt[(size_t)b * NRr + tid];
  __syncthreads();

  // ---- HH = Ht^T * Ht  via V_WMMA_F32_16X16X4_F32 ----
  // 16 tiles of 16x16; wave w owns tiles (i0, j0) and (i0, j0+16) -> shared A operand.
  {
    int t = wave * 2;
    int i0 = (t >> 2) * 16;
    int j0 = (t & 3) * 16;
    int colA = i0 + (lane & 15);
    int colB0 = j0 + (lane & 15);
    int colB1 = colB0 + 16;
    int koff = (lane >> 4) * 2;  // upper half-wave holds K+2, K+3
    v8f acc0 = {};
    v8f acc1 = {};
    for (int k = 0; k < NRr; k += 4) {
      int r = k + koff;
      v2f a, b0, b1;
      a.x  = sHt[r * HT_STRIDE + colA];
      a.y  = sHt[(r + 1) * HT_STRIDE + colA];
      b0.x = sHt[r * HT_STRIDE + colB0];
      b0.y = sHt[(r + 1) * HT_STRIDE + colB0];
      b1.x = sHt[r * HT_STRIDE + colB1];
      b1.y = sHt[(r + 1) * HT_STRIDE + colB1];
      acc0 = __builtin_amdgcn_wmma_f32_16x16x4_f32(false, a, false, b0,
                                                   (short)0, acc0, false, false);
      acc1 = __builtin_amdgcn_wmma_f32_16x16x4_f32(false, a, false, b1,
                                                   (short)0, acc1, false, false);
    }
    // C layout: VGPR v -> row i0+v (lanes 0-15) / i0+v+8 (lanes 16-31), col j0+(lane&15)
    int rowBase = i0 + (lane >> 4) * 8;
    int cc = colB0;
    #pragma unroll
    for (int v = 0; v < 8; ++v) {
      sHH[(rowBase + v) * HH_STRIDE + cc]      = acc0[v];
      sHH[(rowBase + v) * HH_STRIDE + cc + 16] = acc1[v];
    }
  }

  // ---- yH[i] = sum_r yt[r] * Ht[r][i]  (threads 0..63, in-register) ----
  float yH = 0.0f;
  if (tid < NTt) {
    #pragma unroll 8
    for (int r = 0; r < NRr; ++r) yH = fmaf(sYt[r], sHt[r * HT_STRIDE + tid], yH);
  }
  __syncthreads();  // sHH visible to all

  const float sig = sigmat0[b];
  const float sig2 = sig * sig;

  // ---- 64 unrolled CMD iterations; c = log(1/alpha - 1) = 0 for alpha = 0.5 ----
  float s = 0.0f;
  for (int it = 0; it < NITER; ++it) {
    float ta = fabsf(taui[it]);
    float dl = delta[it];
    float factor = (it == 0) ? 1.0f : ta;
    float xt = fast_tanh(s * 0.5f * factor);
    if (tid < NTt) sXt[tid] = xt;
    __syncthreads();
    float xhh = 0.0f;
    if (tid < NTt) {
      #pragma unroll 8
      for (int i = 0; i < NTt; ++i) xhh = fmaf(sXt[i], sHH[i * HH_STRIDE + tid], xhh);
    }
    __syncthreads();
    float gx = 0.5f * ta * (1.0f - xt * xt);
    float gl = fmaf(gx, xhh - yH, sig2 * fast_tanh(s * 0.5f));
    s = fmaf(-dl, gl, s);
  }

  // ---- output mapping: ft = [(1+xt)/2, (1-xt)/2], then xt ----
  float taL = fabsf(taui[NITER]);
  float xt = fast_tanh(s * 0.5f * taL);
  if (tid < NTt) {
    size_t fbase = ((size_t)b * NTt + tid) * 2;
    out[fbase]     = (1.0f + xt) * 0.5f;
    out[fbase + 1] = (1.0f - xt) * 0.5f;
    out[(size_t)B * NTt * 2 + (size_t)b * NTt + tid] = xt;
  }
}

extern "C" void kernel_launch(void* const* d_in, const int* in_sizes, int n_in,
                              void* d_out, int out_size, void* d_ws, size_t ws_size,
                              hipStream_t stream) {
  const float* yt      = (const float*)d_in[0];
  const float* Ht      = (const float*)d_in[1];
  const float* sigmat0 = (const float*)d_in[2];
  const float* taui    = (const float*)d_in[3];
  const float* delta   = (const float*)d_in[4];
  float* out = (float*)d_out;
  int B = in_sizes[2];  // sigmat0 has one entry per batch
  cmdnet_kernel<<<B, 256, 0, stream>>>(yt, Ht, sigmat0, taui, delta, out, B);
}